// ConvLayer_67293547594211
// MI455X (gfx1250) — compile-verified
//
#include <hip/hip_runtime.h>

typedef __attribute__((ext_vector_type(2))) float v2f;
typedef __attribute__((ext_vector_type(4))) float v4f;
typedef __attribute__((ext_vector_type(8))) float v8f;

#define N_ATOMS 100000
#define MNB     12
#define AF      64
#define BF      41
#define FIN     169          // 2A + B
#define KPAD    172          // FIN padded to multiple of 4
#define NC      128          // 2A output columns
#define NM      (N_ATOMS * MNB)
#define BN_EPS  1e-5f

// dynamic LDS layout (floats): WT[128*172] | A[4*16*172] | stats[256]
#define SM_WT    0
#define SM_A     (NC * KPAD)
#define SM_STATS (SM_A + 4 * 16 * KPAD)
#define SM_FLOATS (SM_STATS + 2 * NC)
#define SM_BYTES  (SM_FLOATS * 4)          // 133120 B (< 320KB/WG on CDNA5)

// ---------------------------------------------------------------------------
// Pass 0: transpose W [169,128] -> WT [128][172] (zero padded K) in workspace
// ---------------------------------------------------------------------------
__global__ void wt_transpose_kernel(const float* __restrict__ W, float* __restrict__ wt) {
    int i = blockIdx.x * blockDim.x + threadIdx.x;
    if (i >= NC * KPAD) return;
    int k = i % KPAD;
    int j = i / KPAD;
    wt[i] = (k < FIN) ? W[k * NC + j] : 0.0f;
}

__device__ __forceinline__ float softplus_f(float x) {
    return fmaxf(x, 0.0f) + log1pf(__expf(-fabsf(x)));
}

// ---------------------------------------------------------------------------
// GEMM kernel. 128 threads = 4 waves; each wave handles 48 consecutive rows
// (= exactly 4 atoms) as three 16-row WMMA tiles of the full 128 columns.
// B matrix (WT) lives in LDS, shared by all 4 waves.
// PASS==0: accumulate per-column sum/sumsq of (total@W + b)   -> stats1
// PASS==1: apply BN1 + sigmoid*softplus, reduce over M, store nbr_sumed->out
// ---------------------------------------------------------------------------
template <int PASS>
__global__ void __launch_bounds__(128, 2)
conv_gemm_kernel(const float* __restrict__ atom,
                 const float* __restrict__ nbr,
                 const int*   __restrict__ idx,
                 const float* __restrict__ wt,      // WT[128][KPAD]
                 const float* __restrict__ bias,
                 const float* __restrict__ gamma1,
                 const float* __restrict__ beta1,
                 float*       __restrict__ gstats,  // [256] sum|sumsq
                 float*       __restrict__ out)     // nbr_sumed [N,64]
{
    extern __shared__ float smem[];
    float* s_wt    = smem + SM_WT;
    float* s_a     = smem + SM_A;
    float* s_stats = smem + SM_STATS;

    const int lane = threadIdx.x & 31;
    const int w    = threadIdx.x >> 5;     // wave id 0..3
    const int lo   = lane & 15;
    const int hi   = lane >> 4;            // 0: K+0/K+1 rows M=v, 1: K+2/K+3 rows M=v+8

    // ---- stage WT into LDS (coalesced 16B copies) ----
    for (int i = threadIdx.x; i < (NC * KPAD) / 4; i += blockDim.x) {
        ((v4f*)s_wt)[i] = ((const v4f*)wt)[i];
    }
    if (PASS == 0) {
        for (int i = threadIdx.x; i < 2 * NC; i += blockDim.x) s_stats[i] = 0.0f;
    }

    // Pass 1: preload folded BN1 scale/shift per column j = 16c + lo
    float sc[8], sh[8];
    if (PASS == 1) {
        const float inv = 1.0f / (float)NM;
#pragma unroll
        for (int c = 0; c < 8; ++c) {
            int j = c * 16 + lo;
            float mean = gstats[j] * inv;
            float var  = gstats[NC + j] * inv - mean * mean;
            float s    = rsqrtf(var + BN_EPS) * gamma1[j];
            sc[c] = s;
            sh[c] = (bias[j] - mean) * s + beta1[j];
        }
    }

    float* At = s_a + w * 16 * KPAD;
    const int waveRow0 = blockIdx.x * 192 + w * 48;   // 48 rows = 4 atoms
    float asum[4][4];                                  // [local atom][col tile]
#pragma unroll
    for (int la = 0; la < 4; ++la)
#pragma unroll
        for (int c = 0; c < 4; ++c) asum[la][c] = 0.0f;

    __syncthreads();   // WT + stats ready

#pragma unroll
    for (int t = 0; t < 3; ++t) {
        const int rbase = waveRow0 + t * 16;

        // ---- stage A tile [16][KPAD] into LDS (central | neighbor | edge) ----
        for (int rr = 0; rr < 16; ++rr) {
            int r  = rbase + rr;
            int n  = r / MNB;
            int nb = idx[r];
            const float* cen = atom + (size_t)n  * AF;
            const float* nei = atom + (size_t)nb * AF;
            const float* nf  = nbr  + (size_t)r  * BF;
            for (int j = lane; j < KPAD; j += 32) {
                float v;
                if      (j < AF)      v = cen[j];
                else if (j < 2 * AF)  v = nei[j - AF];
                else if (j < FIN)     v = nf[j - 2 * AF];
                else                  v = 0.0f;
                At[rr * KPAD + j] = v;
            }
        }
        __syncthreads();

        // ---- 16x128 WMMA GEMM, K = 172 in steps of 4 ----
        v8f acc[8];
#pragma unroll
        for (int c = 0; c < 8; ++c) acc[c] = (v8f){0, 0, 0, 0, 0, 0, 0, 0};

        const float* arow = At   + lo * KPAD;      // A: M = lo on lanes
        const float* brow = s_wt + lo * KPAD;      // B: N = 16c + lo on lanes
        // unroll 2 => loads merge into ds_load_2addr_b64; loop stays rolled so
        // the scheduler cannot interchange c/ks (which caused scratch spills)
#pragma unroll 2
        for (int ks = 0; ks < KPAD / 4; ++ks) {
            const int k = 4 * ks + 2 * hi;         // K pair per lane half
            v2f afrag = *(const v2f*)(arow + k);
            v2f bfr[8];
#pragma unroll
            for (int c = 0; c < 8; ++c)
                bfr[c] = *(const v2f*)(brow + c * 16 * KPAD + k);
#pragma unroll
            for (int c = 0; c < 8; ++c)
                acc[c] = __builtin_amdgcn_wmma_f32_16x16x4_f32(
                    false, afrag, false, bfr[c], (short)0, acc[c], false, false);
        }

        if (PASS == 0) {
            // accumulate column sum / sumsq of gated (incl. bias)
#pragma unroll
            for (int c = 0; c < 8; ++c) {
                float bb = bias[c * 16 + lo];
                float s = 0.0f, q = 0.0f;
#pragma unroll
                for (int v = 0; v < 8; ++v) {
                    float g = acc[c][v] + bb;
                    s += g;
                    q += g * g;
                }
                s += __shfl_xor(s, 16, 32);
                q += __shfl_xor(q, 16, 32);
                if (hi == 0) {
                    atomicAdd(&s_stats[c * 16 + lo], s);
                    atomicAdd(&s_stats[NC + c * 16 + lo], q);
                }
            }
        } else {
            // BN1 + sigmoid(filter)*softplus(core), reduce over neighbors
#pragma unroll
            for (int c = 0; c < 4; ++c) {
#pragma unroll
                for (int v = 0; v < 8; ++v) {
                    float gf = acc[c][v]     * sc[c]     + sh[c];
                    float gc = acc[c + 4][v] * sc[c + 4] + sh[c + 4];
                    float f  = 1.0f / (1.0f + __expf(-gf));
                    float p  = softplus_f(gc);
                    float val = f * p;
                    const int la0 = (t * 16 + v) / MNB;       // row for hi==0
                    const int la1 = (t * 16 + v + 8) / MNB;   // row for hi==1
                    if (la0 == la1) {
                        asum[la0][c] += val;
                    } else {
                        asum[la0][c] += hi ? 0.0f : val;
                        asum[la1][c] += hi ? val : 0.0f;
                    }
                }
            }
        }
        __syncthreads();
    }

    if (PASS == 0) {
        for (int i = threadIdx.x; i < 2 * NC; i += blockDim.x)
            atomicAdd(&gstats[i], s_stats[i]);
    } else {
        const int atomBase = blockIdx.x * 16 + w * 4;
#pragma unroll
        for (int la = 0; la < 4; ++la) {
#pragma unroll
            for (int c = 0; c < 4; ++c) {
                float s = asum[la][c] + __shfl_xor(asum[la][c], 16, 32);
                if (hi == 0)
                    out[(size_t)(atomBase + la) * AF + c * 16 + lo] = s;
            }
        }
    }
}

// ---------------------------------------------------------------------------
// Pass 3: column sum/sumsq of nbr_sumed [N,64] for BN2
// ---------------------------------------------------------------------------
__global__ void stats2_kernel(const float* __restrict__ x,
                              float* __restrict__ gsum,
                              float* __restrict__ gsumsq) {
    __shared__ float rs[256];
    __shared__ float rq[256];
    const int col = threadIdx.x & 63;
    const int rg  = threadIdx.x >> 6;   // 0..3 row groups
    float s = 0.0f, q = 0.0f;
    for (int r = blockIdx.x * 4 + rg; r < N_ATOMS; r += gridDim.x * 4) {
        float v = x[(size_t)r * AF + col];
        s += v;
        q += v * v;
    }
    rs[threadIdx.x] = s;
    rq[threadIdx.x] = q;
    __syncthreads();
    if (threadIdx.x < 64) {
        s = rs[col] + rs[col + 64] + rs[col + 128] + rs[col + 192];
        q = rq[col] + rq[col + 64] + rq[col + 128] + rq[col + 192];
        atomicAdd(&gsum[col], s);
        atomicAdd(&gsumsq[col], q);
    }
}

// ---------------------------------------------------------------------------
// Pass 4: out = softplus(atom_in + BN2(nbr_sumed))   (in-place on d_out)
// ---------------------------------------------------------------------------
__global__ void finalize_kernel(const float* __restrict__ atom,
                                const float* __restrict__ gsum,
                                const float* __restrict__ gsumsq,
                                const float* __restrict__ gamma2,
                                const float* __restrict__ beta2,
                                float* __restrict__ out) {
    int i = blockIdx.x * blockDim.x + threadIdx.x;
    if (i >= N_ATOMS * AF) return;
    int col = i & 63;
    const float inv = 1.0f / (float)N_ATOMS;
    float mean = gsum[col] * inv;
    float var  = gsumsq[col] * inv - mean * mean;
    float s    = rsqrtf(var + BN_EPS) * gamma2[col];
    float x    = atom[i] + (out[i] - mean) * s + beta2[col];
    out[i] = softplus_f(x);
}

// ---------------------------------------------------------------------------
extern "C" void kernel_launch(void* const* d_in, const int* in_sizes, int n_in,
                              void* d_out, int out_size, void* d_ws, size_t ws_size,
                              hipStream_t stream) {
    const float* atom   = (const float*)d_in[0];  // [N,64]
    const float* nbr    = (const float*)d_in[1];  // [N,12,41]
    const int*   idx    = (const int*)  d_in[2];  // [N,12]
    const float* W      = (const float*)d_in[3];  // [169,128]
    const float* bias   = (const float*)d_in[4];  // [128]
    const float* gamma1 = (const float*)d_in[5];
    const float* beta1  = (const float*)d_in[6];
    const float* gamma2 = (const float*)d_in[7];
    const float* beta2  = (const float*)d_in[8];
    float* out = (float*)d_out;                   // [N,64]

    float* ws      = (float*)d_ws;
    float* stats1  = ws;            // [0..255]  sum1|sumsq1
    float* sum2    = ws + 256;      // [256..319]
    float* sumsq2  = ws + 320;      // [320..383]
    float* wt      = ws + 384;      // WT[128][172] = 22016 floats

    hipMemsetAsync(ws, 0, 384 * sizeof(float), stream);

    wt_transpose_kernel<<<(NC * KPAD + 255) / 256, 256, 0, stream>>>(W, wt);

    const int gemmBlocks = NM / 192;   // 6250, 192 rows (16 atoms) per block
    conv_gemm_kernel<0><<<gemmBlocks, 128, SM_BYTES, stream>>>(
        atom, nbr, idx, wt, bias, gamma1, beta1, stats1, out);
    conv_gemm_kernel<1><<<gemmBlocks, 128, SM_BYTES, stream>>>(
        atom, nbr, idx, wt, bias, gamma1, beta1, stats1, out);

    stats2_kernel<<<250, 256, 0, stream>>>(out, sum2, sumsq2);

    finalize_kernel<<<(N_ATOMS * AF + 255) / 256, 256, 0, stream>>>(
        atom, sum2, sumsq2, gamma2, beta2, out);
}